// CNFLayer_14139032338989
// MI455X (gfx1250) — compile-verified
//
#include <hip/hip_runtime.h>

// CNF GNN layer for MI455X (gfx1250, wave32).
//
// Pipeline (bandwidth-bound: ~3.3 GB edge traffic >> ~6.6 GFLOP of GEMM):
//   1) Wh = lit_feat @ W_l2c + b_l2c           (WMMA f32 16x16x4, LDS-staged A)
//   2) scatter Wh rows over edges into clause accumulators (f32 atomics)
//   3) cembs = relu(acc / max(deg,1))          (in place, float4)
//   4) scatter cembs rows + clause_feat scalar + degree into literal accs
//   5) out = relu((acc@W0 + s*w128 + d*b) / max(d,1))  (WMMA f32 + epilogue)
// Steps 4/5 use linearity of the second GEMM to push it after the segment
// sum: 100k rows instead of 400k, and no 205MB Wh_c2l intermediate.
//
// CDNA5 specifics: v_wmma_f32_16x16x4_f32 for the f32 GEMMs;
// global_load_async_to_lds_b128 + s_wait_asynccnt to stage the shared A tile
// once per block (8 waves reuse it from LDS instead of 8x redundant global
// reads); hardware global_atomic_add_f32 for the segment sums.

typedef __attribute__((ext_vector_type(2))) float v2f;
typedef __attribute__((ext_vector_type(8))) float v8f;

#define FEAT 128   // IN_SIZE == CLAUSE_SIZE == OUT_SIZE == 128
#define APAD 132   // padded LDS row stride (floats): 16B-aligned rows,
                   // banks = 4*m mod 64 -> conflict-free b64 reads

__device__ __forceinline__ void atomic_add_f32(float* p, float v) {
  // relaxed agent-scope fp32 add -> global_atomic_add_f32 (no CAS loop)
  __hip_atomic_fetch_add(p, v, __ATOMIC_RELAXED, __HIP_MEMORY_SCOPE_AGENT);
}

// Zero-fill, float4 main body + scalar tail. Threads = n/4 + (n&3).
__global__ void k_zero(float* __restrict__ p, size_t n) {
  const size_t n4 = n >> 2;
  const size_t i = (size_t)blockIdx.x * blockDim.x + threadIdx.x;
  if (i < n4) {
    ((float4*)p)[i] = make_float4(0.f, 0.f, 0.f, 0.f);
  } else if (i - n4 < (n & 3)) {
    p[(n4 << 2) + (i - n4)] = 0.f;
  }
}

// ---------------------------------------------------------------------------
// Stage a 16x128 A tile into LDS, once per block, using the CDNA5 async
// LDS-load path. 512 x 16B chunks, 2 per thread. Each wave waits its own
// ASYNCcnt, then the block barrier makes the tile visible to all 8 waves.
// ---------------------------------------------------------------------------
__device__ __forceinline__ void stage_a_tile(const float* __restrict__ A,
                                             float (&As)[16 * APAD],
                                             int m0, int M) {
  const int tid = threadIdx.x;
#pragma unroll
  for (int c = tid; c < 512; c += 256) {
    const int row   = c >> 5;       // 0..15
    const int chunk = c & 31;       // 16B chunk within the 512B row
    const int grow  = min(m0 + row, M - 1);
    const float* g  = A + (size_t)grow * FEAT + chunk * 4;
    // addrspace(3)->generic cast keeps the LDS offset in the low 32 bits
    const unsigned lds = (unsigned)(size_t)(void*)&As[row * APAD + chunk * 4];
    asm volatile("global_load_async_to_lds_b128 %0, %1, off"
                 :: "v"(lds), "v"(g)
                 : "memory");
  }
  asm volatile("s_wait_asynccnt 0x0" ::: "memory");
  __syncthreads();
}

// ---------------------------------------------------------------------------
// GEMM1: out[M,128] = A[M,128] @ W[128,128] + bias[128]
// Block = 256 threads = 8 waves. Block b owns rows [16b,16b+16);
// wave w owns cols [16w,16w+16). K-loop: 32 x v_wmma_f32_16x16x4_f32.
//
// Fragment layouts (ISA 7.12.2, 32-bit, wave32):
//   A 16x4 : lane L -> M=L&15;  half (L>>4) selects K pair {0,1}/{2,3};
//            vgpr 0/1 = K within pair  -> contiguous float2 per lane.
//   B 4x16 : lane L -> N=L&15;  half selects K pair; vgpr = K within pair.
//   C 16x16: vgpr r, lanes 0-15 -> M=r, lanes 16-31 -> M=r+8; N = L&15.
// ---------------------------------------------------------------------------
__global__ void __launch_bounds__(256)
k_gemm_l2c(const float* __restrict__ A, const float* __restrict__ W,
           const float* __restrict__ bias, float* __restrict__ out, int M) {
  __shared__ float As[16 * APAD];
  const int lane = threadIdx.x & 31;
  const int wave = threadIdx.x >> 5;
  const int m0 = blockIdx.x * 16;
  const int n0 = wave * 16;
  const int mr = lane & 15;
  const int kh = (lane >> 4) << 1;                 // 0 or 2

  stage_a_tile(A, As, m0, M);

  const float* __restrict__ Ap = &As[mr * APAD];
  v8f acc = {0.f, 0.f, 0.f, 0.f, 0.f, 0.f, 0.f, 0.f};
#pragma unroll 4
  for (int k = 0; k < FEAT; k += 4) {
    v2f a = *(const v2f*)(Ap + k + kh);            // ds_load_b64
    v2f b;
    b.x = W[(size_t)(k + kh) * FEAT + n0 + mr];    // B[k+kh][n]
    b.y = W[(size_t)(k + kh + 1) * FEAT + n0 + mr];
    acc = __builtin_amdgcn_wmma_f32_16x16x4_f32(
        /*neg_a=*/false, a, /*neg_b=*/false, b,
        /*c_mod=*/(short)0, acc, /*reuse_a=*/false, /*reuse_b=*/false);
  }

  const int cn = n0 + mr;
  const int cm = (lane >> 4) << 3;                 // 0 or 8
  const float bn = bias[cn];
  float* __restrict__ orow = out + (size_t)(m0 + cm) * FEAT + cn;
  if (m0 + 16 <= M) {                              // uniform fast path
#pragma unroll
    for (int r = 0; r < 8; ++r) orow[(size_t)r * FEAT] = acc[r] + bn;
  } else {
#pragma unroll
    for (int r = 0; r < 8; ++r)
      if (m0 + cm + r < M) orow[(size_t)r * FEAT] = acc[r] + bn;
  }
}

// ---------------------------------------------------------------------------
// GEMM2 + c2l epilogue:
//   v   = acc_l[M,128] @ W[0:128,:] + sfeat[m]*W[128,:] + deg[m]*bias
//   out = relu(v / max(deg,1))
// ---------------------------------------------------------------------------
__global__ void __launch_bounds__(256)
k_gemm_c2l(const float* __restrict__ A, const float* __restrict__ sfeat,
           const float* __restrict__ deg, const float* __restrict__ W,
           const float* __restrict__ bias, float* __restrict__ out, int M) {
  __shared__ float As[16 * APAD];
  const int lane = threadIdx.x & 31;
  const int wave = threadIdx.x >> 5;
  const int m0 = blockIdx.x * 16;
  const int n0 = wave * 16;
  const int mr = lane & 15;
  const int kh = (lane >> 4) << 1;

  stage_a_tile(A, As, m0, M);

  const float* __restrict__ Ap = &As[mr * APAD];
  v8f acc = {0.f, 0.f, 0.f, 0.f, 0.f, 0.f, 0.f, 0.f};
#pragma unroll 4
  for (int k = 0; k < FEAT; k += 4) {
    v2f a = *(const v2f*)(Ap + k + kh);
    v2f b;
    b.x = W[(size_t)(k + kh) * FEAT + n0 + mr];
    b.y = W[(size_t)(k + kh + 1) * FEAT + n0 + mr];
    acc = __builtin_amdgcn_wmma_f32_16x16x4_f32(
        false, a, false, b, (short)0, acc, false, false);
  }

  const int cn = n0 + mr;
  const int cm = (lane >> 4) << 3;
  const float w128 = W[(size_t)FEAT * FEAT + cn];  // row 128 = clause_feat wt
  const float bn = bias[cn];
  float* __restrict__ orow = out + (size_t)(m0 + cm) * FEAT + cn;
  if (m0 + 16 <= M) {                              // uniform fast path
#pragma unroll
    for (int r = 0; r < 8; ++r) {
      const int row = m0 + cm + r;
      const float d = deg[row];
      const float v = acc[r] + sfeat[row] * w128 + d * bn;
      orow[(size_t)r * FEAT] = fmaxf(v / fmaxf(d, 1.0f), 0.0f);
    }
  } else {
#pragma unroll
    for (int r = 0; r < 8; ++r) {
      const int row = m0 + cm + r;
      if (row < M) {
        const float d = deg[row];
        const float v = acc[r] + sfeat[row] * w128 + d * bn;
        orow[(size_t)r * FEAT] = fmaxf(v / fmaxf(d, 1.0f), 0.0f);
      }
    }
  }
}

// ---------------------------------------------------------------------------
// Edge scatter, l2c: one wave per edge; lane carries 4 floats (float4 gather,
// 4 x global_atomic_add_f32). Lane 0 bumps the float degree counter.
// ---------------------------------------------------------------------------
__global__ void k_scatter_l2c(const float* __restrict__ Wh, const int* __restrict__ el,
                              const int* __restrict__ ec, float* __restrict__ acc,
                              float* __restrict__ deg, int n_edge) {
  const int e = blockIdx.x * (blockDim.x >> 5) + (threadIdx.x >> 5);
  if (e >= n_edge) return;
  const int lane = threadIdx.x & 31;
  const int src = el[e];   // literal
  const int dst = ec[e];   // clause
  const float4 v = *(const float4*)(Wh + (size_t)src * FEAT + lane * 4);
  float* a = acc + (size_t)dst * FEAT + lane * 4;
  atomic_add_f32(a + 0, v.x);
  atomic_add_f32(a + 1, v.y);
  atomic_add_f32(a + 2, v.z);
  atomic_add_f32(a + 3, v.w);
  if (lane == 0) atomic_add_f32(deg + dst, 1.0f);
}

// Edge scatter, c2l (reverse): cembs row + scalar clause feature + degree.
__global__ void k_scatter_c2l(const float* __restrict__ cembs, const float* __restrict__ cf,
                              const int* __restrict__ el, const int* __restrict__ ec,
                              float* __restrict__ acc, float* __restrict__ sfeat,
                              float* __restrict__ deg, int n_edge) {
  const int e = blockIdx.x * (blockDim.x >> 5) + (threadIdx.x >> 5);
  if (e >= n_edge) return;
  const int lane = threadIdx.x & 31;
  const int src = ec[e];   // clause
  const int dst = el[e];   // literal
  const float4 v = *(const float4*)(cembs + (size_t)src * FEAT + lane * 4);
  float* a = acc + (size_t)dst * FEAT + lane * 4;
  atomic_add_f32(a + 0, v.x);
  atomic_add_f32(a + 1, v.y);
  atomic_add_f32(a + 2, v.z);
  atomic_add_f32(a + 3, v.w);
  if (lane == 0) {
    atomic_add_f32(sfeat + dst, cf[src]);
    atomic_add_f32(deg + dst, 1.0f);
  }
}

// acc[i] = relu(acc[i] / max(deg[row],1)), float4 lanes; 32 float4 per row.
__global__ void k_mean_relu4(float4* __restrict__ acc, const float* __restrict__ deg,
                             size_t n4) {
  const size_t i = (size_t)blockIdx.x * blockDim.x + threadIdx.x;
  if (i >= n4) return;
  const float d = fmaxf(deg[i >> 5], 1.0f);
  float4 v = acc[i];
  v.x = fmaxf(v.x / d, 0.f);
  v.y = fmaxf(v.y / d, 0.f);
  v.z = fmaxf(v.z / d, 0.f);
  v.w = fmaxf(v.w / d, 0.f);
  acc[i] = v;
}

extern "C" void kernel_launch(void* const* d_in, const int* in_sizes, int n_in,
                              void* d_out, int out_size, void* d_ws, size_t ws_size,
                              hipStream_t stream) {
  const float* lit_feat    = (const float*)d_in[0];  // [n_lit, 128]
  const float* clause_feat = (const float*)d_in[1];  // [n_clause, 1]
  const int*   edge_lit    = (const int*)d_in[2];    // [n_edge]
  const int*   edge_clause = (const int*)d_in[3];    // [n_edge]
  const float* W_l2c       = (const float*)d_in[4];  // [128,128]
  const float* b_l2c       = (const float*)d_in[5];  // [128]
  const float* W_c2l       = (const float*)d_in[6];  // [129,128]
  const float* b_c2l       = (const float*)d_in[7];  // [128]
  float* out = (float*)d_out;

  const int n_lit    = in_sizes[0] / FEAT;
  const int n_clause = in_sizes[1];
  const int n_edge   = in_sizes[2];

  // Workspace layout (floats). acc_l reuses the Wh buffer (Wh is dead after
  // the l2c scatter). Total ~260 MB.
  float* Wh    = (float*)d_ws;                         // n_lit*128
  float* acc_c = Wh + (size_t)n_lit * FEAT;            // n_clause*128
  float* deg_c = acc_c + (size_t)n_clause * FEAT;      // n_clause
  float* sfeat = deg_c + n_clause;                     // n_lit
  float* deg_l = sfeat + n_lit;                        // n_lit
  float* acc_l = Wh;                                   // reuse

  const int ewaves = 8;  // edges per 256-thread block

  // 1) zero acc_c, deg_c, sfeat, deg_l (contiguous region)
  {
    size_t z = (size_t)n_clause * FEAT + (size_t)n_clause + 2 * (size_t)n_lit;
    size_t th = (z >> 2) + (z & 3);
    k_zero<<<(unsigned)((th + 255) / 256), 256, 0, stream>>>(acc_c, z);
  }
  // 2) Wh = lit_feat @ W_l2c + b_l2c
  k_gemm_l2c<<<(n_lit + 15) / 16, 256, 0, stream>>>(lit_feat, W_l2c, b_l2c, Wh, n_lit);
  // 3) scatter Wh rows into clause accumulators
  k_scatter_l2c<<<(n_edge + ewaves - 1) / ewaves, 256, 0, stream>>>(
      Wh, edge_lit, edge_clause, acc_c, deg_c, n_edge);
  // 4) cembs = relu(acc_c / max(deg_c,1)) in place
  {
    size_t n4 = (size_t)n_clause * (FEAT / 4);
    k_mean_relu4<<<(unsigned)((n4 + 255) / 256), 256, 0, stream>>>(
        (float4*)acc_c, deg_c, n4);
  }
  // 5) zero literal accumulators (Wh now dead -> reuse)
  {
    size_t t = (size_t)n_lit * FEAT;
    size_t th = t >> 2;  // divisible by 4
    k_zero<<<(unsigned)((th + 255) / 256), 256, 0, stream>>>(acc_l, t);
  }
  // 6) reverse scatter: cembs rows + clause_feat scalar + degree into literals
  k_scatter_c2l<<<(n_edge + ewaves - 1) / ewaves, 256, 0, stream>>>(
      acc_c, clause_feat, edge_lit, edge_clause, acc_l, sfeat, deg_l, n_edge);
  // 7) out = relu((acc_l@W0 + sfeat*w128 + deg*b) / max(deg,1))
  k_gemm_c2l<<<(n_lit + 15) / 16, 256, 0, stream>>>(
      acc_l, sfeat, deg_l, W_c2l, b_c2l, out, n_lit);
}